// TNNCell_4930622456529
// MI455X (gfx1250) — compile-verified
//
#include <hip/hip_runtime.h>
#include <cmath>

typedef float v2f __attribute__((ext_vector_type(2)));
typedef float v8f __attribute__((ext_vector_type(8)));

#define T_STEPS 512
#define FEAT    7

#if defined(__has_builtin)
#if __has_builtin(__builtin_amdgcn_tanhf)
#define HW_TANH 1
#endif
#endif

// tanh: prefer gfx1250 v_tanh_f32; fallback = exp + raw v_rcp_f32 (no IEEE div chain)
__device__ __forceinline__ float fast_tanh(float x) {
#ifdef HW_TANH
    return __builtin_amdgcn_tanhf(x);
#else
    x = fminf(fmaxf(x, -20.0f), 20.0f);
    float e = __expf(2.0f * x);
    return (e - 1.0f) * __builtin_amdgcn_rcpf(e + 1.0f);
#endif
}
// sigmoid(x) = 0.5 + 0.5*tanh(x/2) on the hw-tanh path; else rcp(1+exp(-x))
__device__ __forceinline__ float fast_sigmoid(float x) {
#ifdef HW_TANH
    return fmaf(0.5f, __builtin_amdgcn_tanhf(0.5f * x), 0.5f);
#else
    return __builtin_amdgcn_rcpf(1.0f + __expf(-x));
#endif
}

// One wave (32 lanes) owns 16 batch rows for the full 512-step scan.
// GEMMs use V_WMMA_F32_16X16X4_F32 (f32 in/out, K=4 per instruction).
//   A frag (16x4 f32, 2 VGPRs): lanes 0-15 hold row=lane {K=kb,kb+1},
//                               lanes 16-31 hold row=lane-16 {K=kb+2,kb+3}.
//   B frag (4x16 f32, 2 VGPRs): mirrored: lane%16 = N column, half selects K pair.
//   C/D   (16x16 f32, 8 VGPRs): elem v, lane l -> row = v + 8*(l>=16), col = l%16.
__global__ __launch_bounds__(32) void tnn_scan_kernel(
    const float* __restrict__ in,    // (B, 512, 7)
    const float* __restrict__ caps,  // (1, 4)
    const float* __restrict__ Wg,    // (11, 15)
    const float* __restrict__ bg,    // (15)
    const float* __restrict__ Wp1,   // (11, 16)
    const float* __restrict__ bp1,   // (16)
    const float* __restrict__ Wp2,   // (16, 4)
    const float* __restrict__ bp2,   // (4)
    float* __restrict__ out)         // (B, 512, 4)
{
    __shared__ __align__(16) float allinS[16 * 12]; // all_in rows, K padded 11->12
    __shared__ __align__(16) float gS[16 * 16];     // sigmoid conducts (15 used)
    __shared__ __align__(16) float hS[16 * 16];     // tanh hidden
    __shared__ __align__(16) float pS[16 * 4];      // ploss

    const int lane = threadIdx.x;   // 0..31
    const int hl   = lane >> 4;     // which half of the wave
    const int col  = lane & 15;     // N column / row id inside tile
    const int rowbase = blockIdx.x * 16;

    // ---- loop-invariant B-fragments (weights) ----
    v2f Bg[3], Bh[3], Bp[4];
#pragma unroll
    for (int c = 0; c < 3; ++c) {
        int kb = 4 * c + 2 * hl;
        Bg[c].x = (kb     < 11 && col < 15) ? Wg[(kb    ) * 15 + col] : 0.0f;
        Bg[c].y = (kb + 1 < 11 && col < 15) ? Wg[(kb + 1) * 15 + col] : 0.0f;
        Bh[c].x = (kb     < 11) ? Wp1[(kb    ) * 16 + col] : 0.0f;
        Bh[c].y = (kb + 1 < 11) ? Wp1[(kb + 1) * 16 + col] : 0.0f;
    }
#pragma unroll
    for (int c = 0; c < 4; ++c) {
        int kb = 4 * c + 2 * hl;
        Bp[c].x = (col < 4) ? Wp2[(kb    ) * 4 + col] : 0.0f;
        Bp[c].y = (col < 4) ? Wp2[(kb + 1) * 4 + col] : 0.0f;
    }
    const float bgl  = (col < 15) ? bg[col]  : 0.0f;
    const float bp1l = bp1[col];
    const float bp2l = (col < 4)  ? bp2[col] : 0.0f;

    float ic[4];
#pragma unroll
    for (int i = 0; i < 4; ++i) ic[i] = exp10f(caps[i]);  // inv_caps = 10^caps

    // adjacency index table (compile-time; diagonal masked out)
    static constexpr int Atab[4][6] = {
        {0, 0, 1, 2, 3, 4},
        {0, 0, 5, 6, 7, 8},
        {1, 5, 0, 9, 10, 11},
        {2, 6, 9, 0, 12, 13},
    };

    float p0 = 0.f, p1 = 0.f, p2 = 0.f, p3 = 0.f;  // state (lanes 0-15 authoritative)
    float x5 = 0.f, x6 = 0.f;

    for (int t = 0; t < T_STEPS; ++t) {
        // ---- stage 1: build all_in rows in LDS (lane = row) ----
        if (lane < 16) {
            const float* xp = in + ((size_t)(rowbase + lane) * T_STEPS + t) * FEAT;
            float x0 = xp[0], x1 = xp[1], x2 = xp[2], x3 = xp[3], x4 = xp[4];
            x5 = xp[5]; x6 = xp[6];
            float4* a = (float4*)&allinS[lane * 12];
            a[0] = make_float4(x0, x1, x2, x3);
            a[1] = make_float4(x4, p0, p1, p2);
            a[2] = make_float4(p3, x5, x6, 0.0f);   // K=11 zero pad
        }
        __syncthreads();

        // ---- stage 2: gemm1 (conducts) + gemm2 (hidden), shared A frags ----
        v8f gacc = {};
        v8f hacc = {};
#pragma unroll
        for (int c = 0; c < 3; ++c) {
            int kb = 4 * c + 2 * hl;
            v2f a;
            a.x = allinS[col * 12 + kb];
            a.y = allinS[col * 12 + kb + 1];
            gacc = __builtin_amdgcn_wmma_f32_16x16x4_f32(false, a, false, Bg[c],
                                                         (short)0, gacc, false, false);
            hacc = __builtin_amdgcn_wmma_f32_16x16x4_f32(false, a, false, Bh[c],
                                                         (short)0, hacc, false, false);
        }
        // activations in C-layout, then transpose to LDS
#pragma unroll
        for (int v = 0; v < 8; ++v) {
            int r = hl * 8 + v;
            gS[r * 16 + col] = fast_sigmoid(gacc[v] + bgl);
            hS[r * 16 + col] = fast_tanh(hacc[v] + bp1l);
        }
        __syncthreads();

        // ---- stage 3: gemm3 (ploss), K=16 ----
        v8f pacc = {};
#pragma unroll
        for (int c = 0; c < 4; ++c) {
            int kb = 4 * c + 2 * hl;
            v2f a;
            a.x = hS[col * 16 + kb];
            a.y = hS[col * 16 + kb + 1];
            pacc = __builtin_amdgcn_wmma_f32_16x16x4_f32(false, a, false, Bp[c],
                                                         (short)0, pacc, false, false);
        }
        if (col < 4) {
#pragma unroll
            for (int v = 0; v < 8; ++v) {
                int r = hl * 8 + v;
                pS[r * 4 + col] = fabsf(pacc[v] + bp2l);
            }
        }
        __syncthreads();

        // ---- stage 4: pairwise conductance + state update (lane = row) ----
        if (lane < 16) {
            float gr[16];
            {
                const float4* gsrc = (const float4*)&gS[lane * 16];
                ((float4*)gr)[0] = gsrc[0];
                ((float4*)gr)[1] = gsrc[1];
                ((float4*)gr)[2] = gsrc[2];
                ((float4*)gr)[3] = gsrc[3];
            }
            float4 plv = *(const float4*)&pS[lane * 4];
            float pls[4] = {plv.x, plv.y, plv.z, plv.w};
            float temps[6] = {p0, p1, p2, p3, x5, x6};
            float prevv[4] = {p0, p1, p2, p3};

            // emit output = state BEFORE update (scan collects `prev`)
            *(float4*)&out[((size_t)(rowbase + lane) * T_STEPS + t) * 4] =
                make_float4(p0, p1, p2, p3);

            float nn[4];
#pragma unroll
            for (int i = 0; i < 4; ++i) {
                float td = 0.0f;
#pragma unroll
                for (int j = 0; j < 6; ++j) {
                    if (j == i) continue;                       // mask diagonal
                    td += (temps[j] - prevv[i]) * gr[Atab[i][j]];
                }
                float v = prevv[i] + 0.5f * ic[i] * (td + pls[i]);
                nn[i] = fminf(fmaxf(v, -1.0f), 5.0f);
            }
            p0 = nn[0]; p1 = nn[1]; p2 = nn[2]; p3 = nn[3];
        }
        __syncthreads();  // protect gS/pS reads vs next iteration's writes
    }
}

extern "C" void kernel_launch(void* const* d_in, const int* in_sizes, int n_in,
                              void* d_out, int out_size, void* d_ws, size_t ws_size,
                              hipStream_t stream) {
    const float* in   = (const float*)d_in[0];
    const float* caps = (const float*)d_in[1];
    const float* Wg   = (const float*)d_in[2];
    const float* bg   = (const float*)d_in[3];
    const float* Wp1  = (const float*)d_in[4];
    const float* bp1  = (const float*)d_in[5];
    const float* Wp2  = (const float*)d_in[6];
    const float* bp2  = (const float*)d_in[7];
    float* out = (float*)d_out;

    const int B = in_sizes[0] / (T_STEPS * FEAT);  // 4096
    dim3 grid(B / 16), block(32);
    hipLaunchKernelGGL(tnn_scan_kernel, grid, block, 0, stream,
                       in, caps, Wg, bg, Wp1, bp1, Wp2, bp2, out);
}